// HungarianMatcher_32031866093807
// MI455X (gfx1250) — compile-verified
//
#include <hip/hip_runtime.h>
#include <hip/hip_bf16.h>

// Problem dims
#define BQ 32
#define QN 900
#define TN 300
#define CN 256

typedef __attribute__((ext_vector_type(2))) float v2f;
typedef __attribute__((ext_vector_type(8))) float v8f;

// ---------------------------------------------------------------------------
// Kernel 1: per-(b,q) softmax stats (row max, sum of exp). One wave per row.
// ---------------------------------------------------------------------------
__global__ __launch_bounds__(256) void softmax_stats_kernel(
    const float* __restrict__ logits,
    float* __restrict__ rowMax, float* __restrict__ rowSum)
{
    const int lane = threadIdx.x & 31;
    const int wave = threadIdx.x >> 5;
    const int row  = blockIdx.x * 8 + wave;   // B*Q = 28800 = 3600 blocks * 8 waves

    const float* x = logits + (size_t)row * CN;
    float vals[8];
    float m = -3.402823466e+38f;
#pragma unroll
    for (int e = 0; e < 8; ++e) {
        vals[e] = x[lane + 32 * e];
        m = fmaxf(m, vals[e]);
    }
#pragma unroll
    for (int off = 16; off > 0; off >>= 1)
        m = fmaxf(m, __shfl_xor(m, off, 32));

    float s = 0.0f;
#pragma unroll
    for (int e = 0; e < 8; ++e)
        s += __expf(vals[e] - m);
#pragma unroll
    for (int off = 16; off > 0; off >>= 1)
        s += __shfl_xor(s, off, 32);

    if (lane == 0) {
        rowMax[row] = m;
        rowSum[row] = s;
    }
}

// ---------------------------------------------------------------------------
// Kernel 2: cost matrix. One wave (32 threads) computes a 16q x 16t tile.
// Class cost via exact f32 WMMA GEMM: prob(16xC) * onehot(Cx16), K-steps of 4.
// A 16x4 f32 layout: lane l (lo=l&15, hi=l>>4): VGPR r holds A[lo][2*hi + r].
// B  4x16 f32 layout (mirror): VGPR r holds B[2*hi + r][lo].
// (Any consistent A/B K-permutation is exact here since B is one-hot.)
// C/D layout: lane l, VGPR r -> (q = 8*hi + r, t = lo) within the tile.
// ---------------------------------------------------------------------------
__global__ __launch_bounds__(32) void cost_kernel(
    const float* __restrict__ logits,  const float* __restrict__ pboxes,
    const float* __restrict__ pcutin,  const int*   __restrict__ labels,
    const float* __restrict__ tboxes,  const float* __restrict__ tcutin,
    const float* __restrict__ rowMax,  const float* __restrict__ rowSum,
    float* __restrict__ Cout, float* __restrict__ Ct)
{
    const int lane = threadIdx.x;
    const int lo = lane & 15;
    const int hi = lane >> 4;
    const int q0 = blockIdx.x * 16;
    const int t0 = blockIdx.y * 16;
    const int b  = blockIdx.z;

    // A-side: query row this lane feeds (clamped; tail rows discarded on store)
    const int    qa   = min(q0 + lo, QN - 1);
    const size_t rowA = (size_t)b * QN + qa;
    const float  rmax = rowMax[rowA];
    const float  rinv = 1.0f / rowSum[rowA];
    const float* lrow = logits + rowA * CN + 2 * hi;

    // B-side: one-hot column (target label), clamped for tail
    const int tcl = min(t0 + lo, TN - 1);
    const int lab = labels[b * TN + tcl];
    const int kA0 = 2 * hi;

    v8f acc = {};
#pragma unroll 4
    for (int kb = 0; kb < CN; kb += 4) {
        const v2f l2 = *(const v2f*)(lrow + kb);
        v2f a, bm;
        a.x  = __expf(l2.x - rmax) * rinv;          // prob[qa][kb+2*hi]
        a.y  = __expf(l2.y - rmax) * rinv;          // prob[qa][kb+2*hi+1]
        bm.x = (lab == kb + kA0)     ? 1.0f : 0.0f; // onehot[kb+2*hi][t]
        bm.y = (lab == kb + kA0 + 1) ? 1.0f : 0.0f;
        acc = __builtin_amdgcn_wmma_f32_16x16x4_f32(
                  false, a, false, bm, (short)0, acc, false, false);
    }
    // acc[r] now holds prob[q0+8*hi+r][label[t0+lo]] (exact gather)

    const int t = t0 + lo;
    if (t < TN) {
        const float4 tb = *(const float4*)(tboxes + ((size_t)b * TN + t) * 4);
        const float  tc = tcutin[b * TN + t];
        const float  area_t = (tb.z - tb.x) * (tb.w - tb.y);
#pragma unroll
        for (int r = 0; r < 8; ++r) {
            const int q = q0 + 8 * hi + r;
            if (q < QN) {
                const float4 pb = *(const float4*)(pboxes + ((size_t)b * QN + q) * 4);
                const float  pc = pcutin[b * QN + q];

                const float cbbox = fabsf(pb.x - tb.x) + fabsf(pb.y - tb.y) +
                                    fabsf(pb.z - tb.z) + fabsf(pb.w - tb.w);
                const float area_p = (pb.z - pb.x) * (pb.w - pb.y);
                const float iw  = fminf(pb.z, tb.z) - fmaxf(pb.x, tb.x);
                const float ih  = fminf(pb.w, tb.w) - fmaxf(pb.y, tb.y);
                const float inter = fmaxf(iw, 0.0f) * fmaxf(ih, 0.0f);
                const float uni   = area_p + area_t - inter;
                const float iou   = inter / uni;
                const float ew  = fmaxf(pb.z, tb.z) - fminf(pb.x, tb.x);
                const float eh  = fmaxf(pb.w, tb.w) - fminf(pb.y, tb.y);
                const float enc = fmaxf(ew, 0.0f) * fmaxf(eh, 0.0f);
                const float giou = iou - (enc - uni) / enc;
                const float ccut = fabsf(pc - tc);

                const float cost = 5.0f * cbbox - acc[r] - 2.0f * giou + 2.0f * ccut;
                Cout[((size_t)b * QN + q) * TN + t] = cost;
                Ct  [((size_t)b * TN + t) * QN + q] = cost;  // transposed for LSA
            }
        }
    }
}

// ---------------------------------------------------------------------------
// Kernel 3: Jonker-Volgenant LSA per batch, single wave32 per batch.
// Columns 1..900 are distributed across 32 lanes (29 slots/lane); minv[], v[]
// and the used-bitmask live in registers (fully unrolled, constant indices).
// Argmin is a 5-step shuffle butterfly with smallest-j tie-break (np.argmin).
// Only the random-access state (u, p, way) is in LDS. Single-wave workgroup
// means __syncthreads() lowers to ~nothing (ISA: barrier ops -> S_NOP) and
// acts purely as a compiler LDS fence. Double precision matches the reference.
// ---------------------------------------------------------------------------
#define LSA_N 300
#define LSA_M 900
#define SLOTS 29          // ceil(900/32); slot k of lane l owns column j = l + 32k + 1
#define LSA_INF 1.0e300

__global__ __launch_bounds__(32) void lsa_kernel(
    const float* __restrict__ Ct, float* __restrict__ qidx, float* __restrict__ tidx)
{
    const int b    = blockIdx.x;
    const int lane = threadIdx.x;

    __shared__ double u[LSA_N + 1];
    __shared__ int    p[LSA_M + 1];
    __shared__ int    way[LSA_M + 1];

    double vcol[SLOTS];
    double minv[SLOTS];
    unsigned validM = 0u;
#pragma unroll
    for (int k = 0; k < SLOTS; ++k) {
        vcol[k] = 0.0;
        if (lane + 32 * k < LSA_M) validM |= (1u << k);
    }
    for (int j = lane; j <= LSA_M; j += 32) p[j] = 0;
    for (int i = lane; i <= LSA_N; i += 32) u[i] = 0.0;
    __syncthreads();

    const float* crow_base = Ct + (size_t)b * LSA_N * LSA_M;

    for (int i = 1; i <= LSA_N; ++i) {
        if (lane == 0) p[0] = i;
        unsigned usedM = 0u;
#pragma unroll
        for (int k = 0; k < SLOTS; ++k) minv[k] = LSA_INF;
        __syncthreads();

        int j0 = 0;
        while (true) {
            // mark used[j0] (owner lane sets its register bit; column 0 is virtual)
            if (j0 > 0 && lane == ((j0 - 1) & 31)) usedM |= 1u << ((j0 - 1) >> 5);
            const int    i0   = p[j0];           // LDS broadcast read
            const double ui0  = u[i0];
            const float* crow = crow_base + (size_t)(i0 - 1) * LSA_M;

            double lbest = LSA_INF;
            int    lbj   = LSA_M + 1;
#pragma unroll
            for (int k = 0; k < SLOTS; ++k) {
                const unsigned bit = 1u << k;
                if ((validM & bit) && !(usedM & bit)) {
                    const int c = lane + 32 * k;             // j - 1 (coalesced load)
                    const double cur = (double)crow[c] - ui0 - vcol[k];
                    if (cur < minv[k]) { minv[k] = cur; way[c + 1] = j0; }
                    if (minv[k] < lbest) { lbest = minv[k]; lbj = c + 1; } // ascending j
                }
            }
            // wave-wide argmin (value, then smallest column index on tie)
#pragma unroll
            for (int off = 16; off > 0; off >>= 1) {
                const double ov = __shfl_xor(lbest, off, 32);
                const int    oj = __shfl_xor(lbj,  off, 32);
                if (ov < lbest || (ov == lbest && oj < lbj)) { lbest = ov; lbj = oj; }
            }
            const int    j1    = lbj;
            const double delta = lbest;
            __syncthreads();   // way[] / LDS ordering fence (free: single wave)

            // dual updates: used[0] is always set by now -> u[i] += delta
            if (lane == 0) u[i] += delta;
#pragma unroll
            for (int k = 0; k < SLOTS; ++k) {
                const unsigned bit = 1u << k;
                if (validM & bit) {
                    if (usedM & bit) {
                        vcol[k] -= delta;
                        const int pj = p[lane + 32 * k + 1];
                        u[pj] += delta;          // matched rows distinct -> no conflict
                    } else {
                        minv[k] -= delta;
                    }
                }
            }
            __syncthreads();
            j0 = j1;
            if (p[j0] == 0) break;               // augmenting path reached a free column
        }
        // augment along the alternating path (tiny serial chain)
        if (lane == 0) {
            int jc = j0;
            while (jc) { const int jn = way[jc]; p[jc] = p[jn]; jc = jn; }
        }
        __syncthreads();
    }

    for (int j = 1 + lane; j <= LSA_M; j += 32) {
        const int pi = p[j];
        if (pi > 0) qidx[(size_t)b * LSA_N + (pi - 1)] = (float)(j - 1);
    }
    for (int t = lane; t < LSA_N; t += 32) tidx[(size_t)b * LSA_N + t] = (float)t;
}

// ---------------------------------------------------------------------------
extern "C" void kernel_launch(void* const* d_in, const int* in_sizes, int n_in,
                              void* d_out, int out_size, void* d_ws, size_t ws_size,
                              hipStream_t stream) {
    const float* logits = (const float*)d_in[0];   // (B,Q,C)
    const float* pboxes = (const float*)d_in[1];   // (B,Q,4)
    const float* pcutin = (const float*)d_in[2];   // (B,Q)
    const int*   labels = (const int*)  d_in[3];   // (B,T)
    const float* tboxes = (const float*)d_in[4];   // (B,T,4)
    const float* tcutin = (const float*)d_in[5];   // (B,T)

    float* out  = (float*)d_out;
    float* Cout = out;                                   // B*Q*T
    float* qidx = out + (size_t)BQ * QN * TN;            // B*T
    float* tidx = qidx + (size_t)BQ * TN;                // B*T

    float* rowMax = (float*)d_ws;                        // B*Q
    float* rowSum = rowMax + BQ * QN;                    // B*Q
    float* Ct     = rowSum + BQ * QN;                    // B*T*Q (transposed cost)

    softmax_stats_kernel<<<(BQ * QN) / 8, 256, 0, stream>>>(logits, rowMax, rowSum);

    dim3 grid((QN + 15) / 16, (TN + 15) / 16, BQ);       // 57 x 19 x 32 waves
    cost_kernel<<<grid, 32, 0, stream>>>(logits, pboxes, pcutin, labels,
                                         tboxes, tcutin, rowMax, rowSum, Cout, Ct);

    lsa_kernel<<<BQ, 32, 0, stream>>>(Ct, qidx, tidx);
}